// GraphAttentionLayer_69209103008307
// MI455X (gfx1250) — compile-verified
//
#include <hip/hip_runtime.h>
#include <hip/hip_bf16.h>

// GAT layer: h = x@W^T (WMMA fp32), attention softmax over edges, aggregation.
// N=100000 nodes, E=1600000 edges, IN=128, OUT=64.

#define NN     100000
#define EE     1600000
#define IND    128
#define OUTD   64
#define ALPHA_SLOPE 0.2f
#define EPSV   1e-16f
#define KSTEPS (IND / 4)        // 32 K-steps of 4 for wmma_f32_16x16x4_f32
#define NTILES ((NN + 15) / 16) // 6250 (exact: 100000 = 16*6250)
#define WPAD   130              // 128 + 2 floats pad -> LDS bank-conflict free

typedef __attribute__((ext_vector_type(2))) float v2f;
typedef __attribute__((ext_vector_type(8))) float v8f;

__global__ void gat_zero_kernel(float* __restrict__ p, int n) {
    int i = blockIdx.x * 256 + threadIdx.x;
    if (i < n) p[i] = 0.0f;
}

// One wave per 16-row tile of x; 8 waves (256 threads) per block.
// Computes h tile (16x64) with V_WMMA_F32_16X16X4_F32, then fuses
// s_src = h@a1 and s_dst = h@a2 via cross-lane reduction.
__global__ __launch_bounds__(256) void gat_gemm_kernel(
    const float* __restrict__ x, const float* __restrict__ W,
    const float* __restrict__ attn,
    float* __restrict__ hout, float* __restrict__ s_src, float* __restrict__ s_dst)
{
    __shared__ float sW[OUTD * WPAD]; // 33,280 B: W staged with padded rows

    const int tid = threadIdx.x;
    for (int i = tid; i < OUTD * IND; i += 256) {
        int r = i >> 7;      // / 128
        int k = i & 127;
        sW[r * WPAD + k] = W[i];
    }
    __syncthreads();

    const int wave = tid >> 5;
    const int lane = tid & 31;
    const int tile = blockIdx.x * 8 + wave;
    if (tile >= NTILES) return;          // wave-uniform: EXEC all-1s inside

    const int m0   = tile * 16;
    const int half = lane >> 4;          // 0: K={0,1}, rows M0-7 of C ; 1: K={2,3}, rows M8-15
    const int l15  = lane & 15;
    const int kc   = half * 2;

    const float* xrow = x + (size_t)(m0 + l15) * IND + kc;

    v8f acc0 = {}, acc1 = {}, acc2 = {}, acc3 = {};
    #pragma unroll 4
    for (int s = 0; s < KSTEPS; ++s) {
        v2f a  = *(const v2f*)(xrow + 4 * s);
        v2f b0 = *(const v2f*)(&sW[( 0 + l15) * WPAD + 4 * s + kc]);
        v2f b1 = *(const v2f*)(&sW[(16 + l15) * WPAD + 4 * s + kc]);
        v2f b2 = *(const v2f*)(&sW[(32 + l15) * WPAD + 4 * s + kc]);
        v2f b3 = *(const v2f*)(&sW[(48 + l15) * WPAD + 4 * s + kc]);
        acc0 = __builtin_amdgcn_wmma_f32_16x16x4_f32(false, a, false, b0, (short)0, acc0, false, false);
        acc1 = __builtin_amdgcn_wmma_f32_16x16x4_f32(false, a, false, b1, (short)0, acc1, false, false);
        acc2 = __builtin_amdgcn_wmma_f32_16x16x4_f32(false, a, false, b2, (short)0, acc2, false, false);
        acc3 = __builtin_amdgcn_wmma_f32_16x16x4_f32(false, a, false, b3, (short)0, acc3, false, false);
    }

    // Epilogue: store h tile + fused attention logits.
    const int rowbase = half * 8;
    const float a10 = attn[ 0 + l15], a11 = attn[16 + l15],
                a12 = attn[32 + l15], a13 = attn[48 + l15];
    const float a20 = attn[64 + l15], a21 = attn[80 + l15],
                a22 = attn[96 + l15], a23 = attn[112 + l15];

    float ssrc[8], sdst[8];
    #pragma unroll
    for (int r = 0; r < 8; ++r) {
        float h0 = acc0[r], h1 = acc1[r], h2 = acc2[r], h3 = acc3[r];
        size_t base = (size_t)(m0 + rowbase + r) * OUTD + l15;
        hout[base +  0] = h0;
        hout[base + 16] = h1;
        hout[base + 32] = h2;
        hout[base + 48] = h3;
        float ts = h0 * a10 + h1 * a11 + h2 * a12 + h3 * a13;
        float td = h0 * a20 + h1 * a21 + h2 * a22 + h3 * a23;
        // reduce across the 16 lanes of this half (rows M = rowbase..rowbase+7)
        #pragma unroll
        for (int m = 1; m < 16; m <<= 1) {
            ts += __shfl_xor(ts, m, 32);
            td += __shfl_xor(td, m, 32);
        }
        ssrc[r] = ts;
        sdst[r] = td;
    }
    if (l15 == 0) {
        #pragma unroll
        for (int r = 0; r < 8; ++r) {
            s_src[m0 + rowbase + r] = ssrc[r];
            s_dst[m0 + rowbase + r] = sdst[r];
        }
    }
}

// Pass 1 over edges: e = exp(leaky_relu(s_src[row]+s_dst[col])), segment-sum into e_sum.
__global__ __launch_bounds__(256) void gat_edge_score_kernel(
    const int* __restrict__ row, const int* __restrict__ col,
    const float* __restrict__ s_src, const float* __restrict__ s_dst,
    float* __restrict__ e, float* __restrict__ e_sum)
{
    int i = blockIdx.x * 256 + threadIdx.x;
    if (i >= EE) return;
    int r = row[i], c = col[i];
    float z = s_src[r] + s_dst[c];
    z = z > 0.0f ? z : ALPHA_SLOPE * z;
    float ev = __expf(z);
    e[i] = ev;
    unsafeAtomicAdd(&e_sum[r], ev);   // global_atomic_add_f32 (no return)
}

// Pass 2 over edges: one wave per edge. Lane handles 2 features (float2 gather
// of h[col], two fp32 atomics into out[row]). h/out are L2-resident (25.6 MB each).
__global__ __launch_bounds__(256) void gat_edge_aggr_kernel(
    const int* __restrict__ row, const int* __restrict__ col,
    const float* __restrict__ e, const float* __restrict__ e_sum,
    const float* __restrict__ h, float* __restrict__ out)
{
    int eidx = blockIdx.x * 8 + (threadIdx.x >> 5);
    if (eidx >= EE) return;
    int lane = threadIdx.x & 31;
    int r = row[eidx], c = col[eidx];
    float alpha = e[eidx] / (e_sum[r] + EPSV);
    const float2 hv = *(const float2*)(h + (size_t)c * OUTD + 2 * lane);
    float* op = out + (size_t)r * OUTD + 2 * lane;
    unsafeAtomicAdd(op,     alpha * hv.x);
    unsafeAtomicAdd(op + 1, alpha * hv.y);
}

__global__ void gat_elu_kernel(float* __restrict__ out, int n) {
    int i = blockIdx.x * 256 + threadIdx.x;
    if (i < n) {
        float v = out[i];
        out[i] = v > 0.0f ? v : __expf(v) - 1.0f;
    }
}

extern "C" void kernel_launch(void* const* d_in, const int* in_sizes, int n_in,
                              void* d_out, int out_size, void* d_ws, size_t ws_size,
                              hipStream_t stream)
{
    const float* x    = (const float*)d_in[0];   // [N,128]
    const float* W    = (const float*)d_in[1];   // [64,128]
    const float* attn = (const float*)d_in[2];   // [128,1]
    const int*   ei   = (const int*)d_in[3];     // [2,E]
    const int*   row  = ei;
    const int*   col  = ei + EE;
    float* out = (float*)d_out;                  // [N,64]

    // Workspace layout (fp32): h[N*64] | s_src[N] | s_dst[N] | e_sum[N] | e[E]
    // Total = (6,400,000 + 3*100,000 + 1,600,000) * 4 B ~= 33.2 MB
    float* h_ws   = (float*)d_ws;
    float* s_src  = h_ws  + (size_t)NN * OUTD;
    float* s_dst  = s_src + NN;
    float* e_sum  = s_dst + NN;
    float* e_ws   = e_sum + NN;

    // Zero accumulation targets (harness poisons d_out/d_ws with 0xAA).
    gat_zero_kernel<<<(NN * OUTD + 255) / 256, 256, 0, stream>>>(out, NN * OUTD);
    gat_zero_kernel<<<(NN + 255) / 256, 256, 0, stream>>>(e_sum, NN);

    // h = x @ W^T  (fp32 WMMA) + fused s_src/s_dst.
    gat_gemm_kernel<<<(NTILES + 7) / 8, 256, 0, stream>>>(x, W, attn, h_ws, s_src, s_dst);

    // Edge softmax numerator + denominator.
    gat_edge_score_kernel<<<(EE + 255) / 256, 256, 0, stream>>>(row, col, s_src, s_dst, e_ws, e_sum);

    // Weighted aggregation: out[row] += alpha * h[col].
    gat_edge_aggr_kernel<<<(EE + 7) / 8, 256, 0, stream>>>(row, col, e_ws, e_sum, h_ws, out);

    // Final ELU.
    gat_elu_kernel<<<(NN * OUTD + 255) / 256, 256, 0, stream>>>(out, NN * OUTD);
}